// QConv2dMedium_56856777065114
// MI455X (gfx1250) — compile-verified
//
#include <hip/hip_runtime.h>
#include <math.h>

typedef __attribute__((ext_vector_type(2))) float v2f;
typedef __attribute__((ext_vector_type(8))) float v8f;

#define NIMG 32
#define Hdim 256
#define Wdim 256
#define HW   (Hdim * Wdim)
#define NCH  16

// ---------------------------------------------------------------------------
// Kernel 1: build the 16x16 StronglyEntanglingLayers unitary from weights.
// One wave (32 lanes), 8 matrix elements per lane, staged through LDS.
// Output: G[0..255] = Re(U) row-major [j][k], G[256..511] = Im(U).
// ---------------------------------------------------------------------------
__global__ __launch_bounds__(32)
void build_unitary_kernel(const float* __restrict__ wts, float* __restrict__ G)
{
    __shared__ float Ur[256], Ui[256], Tr[256], Ti[256], Lr[256], Li[256];
    const int lane = threadIdx.x;

#pragma unroll
    for (int e = 0; e < 8; ++e) {
        int idx = lane + 32 * e;
        Ur[idx] = ((idx >> 4) == (idx & 15)) ? 1.0f : 0.0f;
        Ui[idx] = 0.0f;
    }
    __syncthreads();

    for (int l = 0; l < 2; ++l) {
        // qml.Rot(phi, theta, omega) = RZ(omega) RY(theta) RZ(phi), per wire.
        float gr[4][2][2], gi[4][2][2];
#pragma unroll
        for (int i = 0; i < 4; ++i) {
            float phi = wts[l * 12 + i * 3 + 0];
            float th  = wts[l * 12 + i * 3 + 1];
            float om  = wts[l * 12 + i * 3 + 2];
            float c = cosf(0.5f * th), s = sinf(0.5f * th);
            float apo = 0.5f * (phi + om), amo = 0.5f * (phi - om);
            float ca = cosf(apo), sa = sinf(apo);
            float cm = cosf(amo), sm = sinf(amo);
            gr[i][0][0] =  ca * c;  gi[i][0][0] = -sa * c;   //  e^{-i(phi+om)/2} c
            gr[i][0][1] = -cm * s;  gi[i][0][1] = -sm * s;   // -e^{ i(phi-om)/2} s
            gr[i][1][0] =  cm * s;  gi[i][1][0] = -sm * s;   //  e^{-i(phi-om)/2} s
            gr[i][1][1] =  ca * c;  gi[i][1][1] =  sa * c;   //  e^{ i(phi+om)/2} c
        }
        // layer = kron(g0, g1, g2, g3); wire 0 = most significant bit
#pragma unroll
        for (int e = 0; e < 8; ++e) {
            int idx = lane + 32 * e;
            int r = idx >> 4, cc = idx & 15;
            float ar = 1.0f, ai = 0.0f;
#pragma unroll
            for (int i = 0; i < 4; ++i) {
                int rb = (r >> (3 - i)) & 1, cb = (cc >> (3 - i)) & 1;
                float br = gr[i][rb][cb], bi = gi[i][rb][cb];
                float nr = ar * br - ai * bi;
                float ni = ar * bi + ai * br;
                ar = nr; ai = ni;
            }
            Lr[idx] = ar; Li[idx] = ai;
        }
        __syncthreads();
        // T = layer @ U  (complex 16x16 matmul, 8 outputs per lane)
#pragma unroll
        for (int e = 0; e < 8; ++e) {
            int idx = lane + 32 * e;
            int r = idx >> 4, cc = idx & 15;
            float sr = 0.0f, si = 0.0f;
#pragma unroll
            for (int k = 0; k < 16; ++k) {
                float lr = Lr[r * 16 + k], li = Li[r * 16 + k];
                float ur = Ur[k * 16 + cc], ui = Ui[k * 16 + cc];
                sr += lr * ur - li * ui;
                si += lr * ui + li * ur;
            }
            Tr[idx] = sr; Ti[idx] = si;
        }
        __syncthreads();
#pragma unroll
        for (int e = 0; e < 8; ++e) { int idx = lane + 32 * e; Ur[idx] = Tr[idx]; Ui[idx] = Ti[idx]; }
        __syncthreads();

        // CNOT ring: control i -> target (i + (l%3 + 1)) % 4 ; row permutation.
        int rl = (l % 3) + 1;
        for (int i = 0; i < 4; ++i) {
            int tgt  = (i + rl) & 3;
            int cbit = 1 << (3 - i);
            int tbit = 1 << (3 - tgt);
#pragma unroll
            for (int e = 0; e < 8; ++e) {
                int idx = lane + 32 * e;
                int r = idx >> 4, cc = idx & 15;
                int pr = (r & cbit) ? (r ^ tbit) : r;
                Tr[idx] = Ur[pr * 16 + cc];
                Ti[idx] = Ui[pr * 16 + cc];
            }
            __syncthreads();
#pragma unroll
            for (int e = 0; e < 8; ++e) { int idx = lane + 32 * e; Ur[idx] = Tr[idx]; Ui[idx] = Ti[idx]; }
            __syncthreads();
        }
    }
#pragma unroll
    for (int e = 0; e < 8; ++e) {
        int idx = lane + 32 * e;
        G[idx]       = Ur[idx];
        G[256 + idx] = Ui[idx];
    }
}

// ---------------------------------------------------------------------------
// Kernel 2: per wave, process tiles of 16 consecutive patches with fp32 WMMA.
//   A (16x4)  = rows of Re(U)/Im(U)  (M = channel, invariant across tiles)
//   B (4x16)  = normalized patch components (N = patch)
//   D (16x16) = channel x patch; lanes store 16 consecutive floats/channel.
// f32 A/B striping: lanes 0-15 hold K=k0,k0+1; lanes 16-31 hold K=k0+2,k0+3.
// ---------------------------------------------------------------------------
__global__ __launch_bounds__(256)
void qconv_wmma_kernel(const float* __restrict__ x, const float* __restrict__ G,
                       float* __restrict__ out)
{
    const int lane = threadIdx.x & 31;
    const int row  = lane & 15;     // channel row for A, patch column for B/D
    const int hi   = lane >> 4;     // half-wave selector
    const int wavesPerBlock = blockDim.x >> 5;
    const int waveId     = blockIdx.x * wavesPerBlock + (threadIdx.x >> 5);
    const int totalWaves = gridDim.x * wavesPerBlock;
    const int numTiles   = (NIMG * HW) / 16;

    const float* Gre = G;
    const float* Gim = G + 256;
    const int kb = 2 * hi;
    v2f are0, are1, are2, aim0, aim1, aim2;
    are0.x = Gre[row * 16 + 0 + kb];  are0.y = Gre[row * 16 + 1 + kb];
    are1.x = Gre[row * 16 + 4 + kb];  are1.y = Gre[row * 16 + 5 + kb];
    are2.x = Gre[row * 16 + 8 + kb];  are2.y = Gre[row * 16 + 9 + kb];
    aim0.x = Gim[row * 16 + 0 + kb];  aim0.y = Gim[row * 16 + 1 + kb];
    aim1.x = Gim[row * 16 + 4 + kb];  aim1.y = Gim[row * 16 + 5 + kb];
    aim2.x = Gim[row * 16 + 8 + kb];  aim2.y = Gim[row * 16 + 9 + kb];

    for (int t = waveId; t < numTiles; t += totalWaves) {  // wave-uniform: EXEC stays full
        const int base = t << 4;
        const int img  = base / HW;
        const int hw0  = base - img * HW;      // multiple of 16, within one row
        const int h    = hw0 >> 8;
        const int w0   = hw0 & 255;
        const int wq   = w0 + row;             // this lane's patch column
        const float* xi = x + img * HW;

        // 3x3 patch with ConstantPad2d(0.01); clamp addr, select value.
        float p[9];
#pragma unroll
        for (int dy = -1; dy <= 1; ++dy) {
            const int hh = h + dy;
            const int hc = min(max(hh, 0), Hdim - 1);
            const bool hv = (hh == hc);
#pragma unroll
            for (int dx = -1; dx <= 1; ++dx) {
                const int ww = wq + dx;
                const int wc = min(max(ww, 0), Wdim - 1);
                const bool wv = (ww == wc);
                float v = xi[hc * Wdim + wc];
                p[(dy + 1) * 3 + (dx + 1)] = (hv && wv) ? v : 0.01f;
            }
        }
        float s2 = 0.0f;
#pragma unroll
        for (int k = 0; k < 9; ++k) s2 += p[k] * p[k];
        const float inv = 1.0f / fmaxf(sqrtf(s2), 1e-12f);
#pragma unroll
        for (int k = 0; k < 9; ++k) p[k] *= inv;

        // B chunks (K = 0..3, 4..7, 8..11; components 9..15 are zero pad)
        v2f b0, b1, b2;
        b0.x = hi ? p[2] : p[0];  b0.y = hi ? p[3] : p[1];
        b1.x = hi ? p[6] : p[4];  b1.y = hi ? p[7] : p[5];
        b2.x = hi ? 0.0f : p[8];  b2.y = 0.0f;

        v8f z = {};
        v8f dre = __builtin_amdgcn_wmma_f32_16x16x4_f32(false, are0, false, b0, (short)0, z,   false, false);
        dre     = __builtin_amdgcn_wmma_f32_16x16x4_f32(false, are1, false, b1, (short)0, dre, false, false);
        dre     = __builtin_amdgcn_wmma_f32_16x16x4_f32(false, are2, false, b2, (short)0, dre, false, false);
        v8f dim_ = __builtin_amdgcn_wmma_f32_16x16x4_f32(false, aim0, false, b0, (short)0, z,    false, false);
        dim_     = __builtin_amdgcn_wmma_f32_16x16x4_f32(false, aim1, false, b1, (short)0, dim_, false, false);
        dim_     = __builtin_amdgcn_wmma_f32_16x16x4_f32(false, aim2, false, b2, (short)0, dim_, false, false);

        // out[img][c][h][w]: per store, lanes 0-15 / 16-31 each write 16
        // consecutive floats of one channel plane (coalesced 64B segments).
        float* op = out + (size_t)img * NCH * HW + hw0 + row;
        const int c0 = hi << 3;
#pragma unroll
        for (int r = 0; r < 8; ++r) {
            float vr = dre[r], vi = dim_[r];
            float val = fminf(8.0f * (vr * vr + vi * vi), 1.0f);
            op[(size_t)(c0 + r) * HW] = val;
        }
    }
}

// ---------------------------------------------------------------------------
extern "C" void kernel_launch(void* const* d_in, const int* in_sizes, int n_in,
                              void* d_out, int out_size, void* d_ws, size_t ws_size,
                              hipStream_t stream)
{
    const float* x   = (const float*)d_in[0];   // (32,1,256,256) fp32
    const float* wts = (const float*)d_in[1];   // (2,4,3) fp32
    float* out = (float*)d_out;                 // (32,16,256,256) fp32
    float* G   = (float*)d_ws;                  // 512 floats: Re(U), Im(U)

    build_unitary_kernel<<<1, 32, 0, stream>>>(wts, G);
    qconv_wmma_kernel<<<1024, 256, 0, stream>>>(x, G, out);
}